// Attention_52982716563497
// MI455X (gfx1250) — compile-verified
//
#include <hip/hip_runtime.h>

typedef __attribute__((ext_vector_type(2))) float v2f;
typedef __attribute__((ext_vector_type(4))) float v4f;
typedef __attribute__((ext_vector_type(8))) float v8f;
typedef __attribute__((ext_vector_type(4))) int   v4i;

#define NMAX 100000000.0f

static constexpr int Bc = 4, Hc = 16, Sc = 2048, Dc = 64;
static constexpr int KT  = Sc / 16;  // 128 k-tiles
static constexpr int WPB = 8;        // waves (q-tiles) per block
static constexpr int PST = 20;       // LDS P-tile row stride (floats): 16B/8B aligned, conflict-free

__global__ __launch_bounds__(256)
void attn_wmma_f32(const float* __restrict__ Q,
                   const float* __restrict__ K,
                   const float* __restrict__ V,
                   const int*   __restrict__ mask,
                   float* __restrict__ outp,
                   float* __restrict__ pattn)
{
    __shared__ float ptile[WPB][16 * PST];

    const int lane = threadIdx.x & 31;
    const int wave = threadIdx.x >> 5;
    const int lr   = lane & 15;   // q-row within tile
    const int hi   = lane >> 4;   // k-half (0: k=0..7, 1: k=8..15)

    const int bh = blockIdx.y;        // b*H + h
    const int b  = bh >> 4;           // H == 16
    const int q0 = (blockIdx.x * WPB + wave) * 16;

    // Q fragments (used as WMMA *B* operand of S^T = K x Q^T):
    // lane: Q[q0+lr][d0 + 2*hi], Q[q0+lr][d0 + 2*hi + 1]
    const float* qrow = Q + ((size_t)bh * Sc + q0 + lr) * Dc + 2 * hi;
    v2f qf[16];
#pragma unroll
    for (int i = 0; i < 16; ++i)
        qf[i] = *(const v2f*)(qrow + i * 4);

    const float* krow  = K + ((size_t)bh * Sc + lr) * Dc + 2 * hi;
    const int*   mbase = mask + (size_t)b * Sc * Sc + (size_t)(q0 + lr) * Sc + 8 * hi;

    // ---- Pass 1: online softmax stats; lane owns row q0+lr (half of k per lane) ----
    float mrow = -3.0e38f, lrow = 0.0f;

    for (int kt = 0; kt < KT; ++kt) {
        const float* kr = krow + (size_t)kt * 16 * Dc;
        v8f c = {};
#pragma unroll
        for (int i = 0; i < 16; ++i) {
            v2f kf = *(const v2f*)(kr + i * 4);
            // S^T tile: c[r] = score[q0+lr][kt*16 + 8*hi + r]
            c = __builtin_amdgcn_wmma_f32_16x16x4_f32(
                    false, kf, false, qf[i], (short)0, c, false, false);
        }
        v4i mv0 = *(const v4i*)(mbase + kt * 16);
        v4i mv1 = *(const v4i*)(mbase + kt * 16 + 4);
        float s[8];
#pragma unroll
        for (int j = 0; j < 4; ++j) {
            s[j]     = (mv0[j] == 1) ? -NMAX : c[j] * 0.125f;
            s[j + 4] = (mv1[j] == 1) ? -NMAX : c[j + 4] * 0.125f;
        }
        float tm = s[0];
#pragma unroll
        for (int j = 1; j < 8; ++j) tm = fmaxf(tm, s[j]);
        tm = fmaxf(tm, __shfl_xor(tm, 16, 32));      // combine the two k-halves
        float mnew = fmaxf(mrow, tm);
        float e = 0.0f;
#pragma unroll
        for (int j = 0; j < 8; ++j) e += __expf(s[j] - mnew);
        e += __shfl_xor(e, 16, 32);
        lrow = lrow * __expf(mrow - mnew) + e;
        mrow = mnew;
    }

    const float invl = 1.0f / lrow;

    // ---- Pass 2: recompute scores, emit normalized p, accumulate out = P @ V ----
    v8f oacc[4] = {{}, {}, {}, {}};
    float* pg = pattn + ((size_t)bh * Sc + q0 + lr) * Sc + 8 * hi;
    float* tp = &ptile[wave][0];

    for (int kt = 0; kt < KT; ++kt) {
        const float* kr = krow + (size_t)kt * 16 * Dc;
        v8f c = {};
#pragma unroll
        for (int i = 0; i < 16; ++i) {
            v2f kf = *(const v2f*)(kr + i * 4);
            c = __builtin_amdgcn_wmma_f32_16x16x4_f32(
                    false, kf, false, qf[i], (short)0, c, false, false);
        }
        v4i mv0 = *(const v4i*)(mbase + kt * 16);
        v4i mv1 = *(const v4i*)(mbase + kt * 16 + 4);
        float p[8];
#pragma unroll
        for (int j = 0; j < 4; ++j) {
            float s0 = (mv0[j] == 1) ? -NMAX : c[j] * 0.125f;
            float s1 = (mv1[j] == 1) ? -NMAX : c[j + 4] * 0.125f;
            p[j]     = __expf(s0 - mrow) * invl;
            p[j + 4] = __expf(s1 - mrow) * invl;
        }
        v4f pv0 = {p[0], p[1], p[2], p[3]};
        v4f pv1 = {p[4], p[5], p[6], p[7]};
        // final normalized p_attn: row q0+lr, cols kt*16 + 8*hi + 0..7 (32B contiguous)
        *(v4f*)(pg + kt * 16)     = pv0;
        *(v4f*)(pg + kt * 16 + 4) = pv1;
        // stage P[q][k] into LDS for A-fragment reshape
        *(v4f*)(tp + lr * PST + 8 * hi)     = pv0;
        *(v4f*)(tp + lr * PST + 8 * hi + 4) = pv1;
        asm volatile("s_wait_dscnt 0x0" ::: "memory");   // intra-wave LDS store->load fence

        const float* vb = V + ((size_t)bh * Sc + kt * 16 + 2 * hi) * Dc + lr;
#pragma unroll
        for (int kkc = 0; kkc < 4; ++kkc) {
            // A fragment of P: lane -> P[lr][kkc*4 + 2*hi], P[lr][kkc*4 + 2*hi + 1]
            v2f a = *(const v2f*)(tp + lr * PST + kkc * 4 + 2 * hi);
            const float* vr = vb + (size_t)kkc * 4 * Dc;
#pragma unroll
            for (int nc = 0; nc < 4; ++nc) {
                v2f bf;
                bf.x = vr[nc * 16];
                bf.y = vr[(size_t)Dc + nc * 16];
                oacc[nc] = __builtin_amdgcn_wmma_f32_16x16x4_f32(
                               false, a, false, bf, (short)0, oacc[nc], false, false);
            }
        }
        asm volatile("" ::: "memory");   // keep next iter's stores behind these loads
    }

    // ---- Write out [B,H,S,D]: D layout row m = r + 8*hi, col = nc*16 + lr ----
    float* ob = outp + ((size_t)bh * Sc + q0 + 8 * hi) * Dc + lr;
#pragma unroll
    for (int nc = 0; nc < 4; ++nc)
#pragma unroll
        for (int r = 0; r < 8; ++r)
            ob[(size_t)r * Dc + nc * 16] = oacc[nc][r];
}

extern "C" void kernel_launch(void* const* d_in, const int* in_sizes, int n_in,
                              void* d_out, int out_size, void* d_ws, size_t ws_size,
                              hipStream_t stream) {
    const float* Q    = (const float*)d_in[0];
    const float* K    = (const float*)d_in[1];
    const float* V    = (const float*)d_in[2];
    const int*   mask = (const int*)  d_in[3];

    float* outp  = (float*)d_out;
    float* pattn = outp + (size_t)Bc * Hc * Sc * Dc;   // outputs concatenated: out, p_attn

    dim3 grid(Sc / 16 / WPB, Bc * Hc);   // (16, 64)
    dim3 block(32 * WPB);                // 256 threads = 8 waves
    attn_wmma_f32<<<grid, block, 0, stream>>>(Q, K, V, mask, outp, pattn);
}